// CIFAR10Net_86096914416128
// MI455X (gfx1250) — compile-verified
//
#include <hip/hip_runtime.h>
#include <hip/hip_bf16.h>

typedef __attribute__((ext_vector_type(2))) float v2f;
typedef __attribute__((ext_vector_type(8))) float v8f;

#define TT   16      // timesteps
#define TB   32      // batch
#define CIN  3
#define COUT 128
#define HW   32
#define KK   27      // Cin*3*3
#define KP   28      // K padded to multiple of 4

// ---------------------------------------------------------------------------
// Kernel 0: clip + weight-standardize (per-out-channel mean / unbiased var),
// apply norm_weight/norm_bias, write K-padded [COUT][28] fp32 into workspace.
// ---------------------------------------------------------------------------
__global__ __launch_bounds__(128) void prep_weights_kernel(
    const float* __restrict__ w, const float* __restrict__ nw,
    const float* __restrict__ nb, float* __restrict__ wsw) {
  int c = threadIdx.x;
  if (c >= COUT) return;
  float vals[KK];
  float mean = 0.f;
  #pragma unroll
  for (int i = 0; i < KK; ++i) {
    float v = w[c * KK + i];
    v = fminf(4.f, fmaxf(-4.f, v));     // clip to [-4, 4]
    vals[i] = v;
    mean += v;
  }
  mean *= (1.f / 27.f);
  float var = 0.f;
  #pragma unroll
  for (int i = 0; i < KK; ++i) { float d = vals[i] - mean; var += d * d; }
  var *= (1.f / 26.f);                  // ddof=1 (unbiased)
  float scale = rsqrtf(var + 1e-5f) * nw[c];
  float bias  = nb[c];
  #pragma unroll
  for (int i = 0; i < KK; ++i) wsw[c * KP + i] = (vals[i] - mean) * scale + bias;
  wsw[c * KP + KK] = 0.f;               // zero pad row (K=27)
}

// ---------------------------------------------------------------------------
// Kernel 1: fused implicit-GEMM conv (fp32 WMMA 16x16x4) + LIF scan over T.
// Block: 256 threads (8 wave32); wave w -> cout chunk [16w,16w+16);
// block -> (batch b, 16-wide spatial tile). LIF state (u, sg) lives in VGPRs
// across all 16 timesteps; only spikes are written (256 MB, the BW floor).
// ---------------------------------------------------------------------------
__global__ __launch_bounds__(256) void snn_conv_lif_kernel(
    const float* __restrict__ x, const float* __restrict__ wsw,
    const float* __restrict__ threshold, float* __restrict__ out) {
  __shared__ float Bls[KP * 16];        // im2col patch: [k(padded 28)][n(16)]

  const int tid   = threadIdx.x;
  const int lane  = tid & 31;
  const int wave  = tid >> 5;           // 0..7 -> cout chunk
  const int s     = blockIdx.x;         // 0..63 spatial tile
  const int b     = blockIdx.y;         // 0..31 batch
  const int h     = s >> 1;
  const int w0    = (s & 1) << 4;

  const int m     = lane & 15;          // row (A) / col (B) within tile
  const int khalf = lane >> 4;          // K half-select per ISA lane layout
  const int coutBase = wave << 4;

  if (tid < 16) Bls[KK * 16 + tid] = 0.f;   // zero the K=27 pad row once

  // A fragments: step j -> lane holds A[m][4j+2*khalf], A[m][4j+2*khalf+1]
  v2f A[7];
  #pragma unroll
  for (int j = 0; j < 7; ++j) {
    const float* p = wsw + (coutBase + m) * KP + 4 * j + 2 * khalf;
    A[j] = *(const v2f*)p;              // 8B-aligned (KP*4 = 112 bytes/row)
  }

  // Per-element thresholds: C/D VGPR r -> cout = base + r + 8*khalf
  float thr[8];
  #pragma unroll
  for (int r = 0; r < 8; ++r) thr[r] = threshold[coutBase + r + (khalf << 3)];

  v8f u = {};                           // membrane potential
  v8f sg = {};                          // grad-synapse state (sm/ss are dead)
  const float kg = 1.0f - 1.0f / 3.5f;  // (1 - 1/TAU_GRAD)

  for (int t = 0; t < TT; ++t) {
    __syncthreads();                    // WAR: prior iteration done reading LDS
    // Cooperative im2col staging (27*16 = 432 values, 256 threads)
    for (int idx = tid; idx < KK * 16; idx += 256) {
      int k  = idx >> 4, n = idx & 15;
      int ci = k / 9, rem = k - ci * 9;
      int kr = rem / 3, kc = rem - kr * 3;
      int hh = h + kr - 1, ww = w0 + n + kc - 1;
      bool inb = ((unsigned)hh < 32u) && ((unsigned)ww < 32u);
      const float* px = x + (((t * TB + b) * CIN + ci) * HW + hh) * HW + ww;
      float v = inb ? *px : 0.f;
      Bls[idx] = v;
      if (inb && (t + 1 < TT))          // gfx1250 global_prefetch_b8 (next t)
        __builtin_prefetch(px + TB * CIN * HW * HW, 0, 1);
    }
    __syncthreads();

    // acc = W(16x28) x patch(28x16), fp32 WMMA, K=4 per issue
    v8f acc = {};
    #pragma unroll
    for (int j = 0; j < 7; ++j) {
      int krow = 4 * j + 2 * khalf;     // B: VGPR0 rows {4j,4j+2}, VGPR1 {4j+1,4j+3}
      v2f Bf;
      Bf.x = Bls[krow * 16 + m];
      Bf.y = Bls[(krow + 1) * 16 + m];
      acc = __builtin_amdgcn_wmma_f32_16x16x4_f32(
          /*neg_a=*/false, A[j], /*neg_b=*/false, Bf,
          /*c_mod=*/(short)0, acc, /*reuse_a=*/false, /*reuse_b=*/false);
    }

    // LIF update + spike store (reset-on-spike; non-leaky u accumulation)
    #pragma unroll
    for (int r = 0; r < 8; ++r) {
      float g  = (sg[r] + acc[r]) * kg;
      float uu = u[r] + g;
      float spike = (uu >= thr[r]) ? 1.f : 0.f;
      float keep  = 1.f - spike;
      u[r]  = uu * keep;
      sg[r] = g * keep;
      int cout = coutBase + r + (khalf << 3);
      out[(((t * TB + b) * COUT + cout) * HW + h) * HW + (w0 + m)] = spike;
    }
  }
}

// ---------------------------------------------------------------------------
extern "C" void kernel_launch(void* const* d_in, const int* in_sizes, int n_in,
                              void* d_out, int out_size, void* d_ws, size_t ws_size,
                              hipStream_t stream) {
  const float* x      = (const float*)d_in[0];  // [16,32,3,32,32]
  const float* weight = (const float*)d_in[1];  // [128,3,3,3]
  const float* nw     = (const float*)d_in[2];  // [128]
  const float* nb     = (const float*)d_in[3];  // [128]
  const float* th     = (const float*)d_in[4];  // [128]
  float* out = (float*)d_out;                   // [16,32,128,32,32]
  float* wsw = (float*)d_ws;                    // [128][28] standardized weights

  prep_weights_kernel<<<1, 128, 0, stream>>>(weight, nw, nb, wsw);
  snn_conv_lif_kernel<<<dim3(64, 32), 256, 0, stream>>>(x, wsw, th, out);
}